// StochasticTwoLayerGCN_76158360092834
// MI455X (gfx1250) — compile-verified
//
#include <hip/hip_runtime.h>
#include <hip/hip_bf16.h>

typedef __attribute__((ext_vector_type(2))) float v2f;
typedef __attribute__((ext_vector_type(8))) float v8f;

#define D_FEAT 256
#define LDS_STRIDE 260   // 256 + 4 pad: row step = 4 banks -> conflict-free 16-lane column reads

// ---------------------------------------------------------------------------
// Degree accumulation: deg_src[src[e]] += 1, deg_dst[dst[e]] += 1
// ---------------------------------------------------------------------------
__global__ void degree_kernel(const int* __restrict__ src, const int* __restrict__ dst,
                              float* __restrict__ deg_src, float* __restrict__ deg_dst, int E) {
    int e = blockIdx.x * blockDim.x + threadIdx.x;
    if (e < E) {
        atomicAdd(&deg_src[src[e]], 1.0f);
        atomicAdd(&deg_dst[dst[e]], 1.0f);
    }
}

// In-place deg -> rsqrt(max(deg, 1))
__global__ void rsqrt_deg_kernel(float* __restrict__ deg, int n) {
    int i = blockIdx.x * blockDim.x + threadIdx.x;
    if (i < n) deg[i] = rsqrtf(fmaxf(deg[i], 1.0f));
}

// ---------------------------------------------------------------------------
// Edge aggregation: agg[dst[e], :] += feat[src[e], :] * rdeg_src[src[e]]
// One thread per (edge, 4-feature quad): 64 threads cover one 256-f32 row,
// so a 64-thread span reads 1KB contiguous (coalesced) and issues
// global_atomic_add_f32 into the L2-resident aggregation buffer.
// ---------------------------------------------------------------------------
__global__ void aggregate_kernel(const float* __restrict__ feat, const int* __restrict__ src,
                                 const int* __restrict__ dst, const float* __restrict__ rdeg_src,
                                 float* __restrict__ agg, int E) {
    int gid = blockIdx.x * blockDim.x + threadIdx.x;
    int e = gid >> 6;
    if (e >= E) return;
    int q = (gid & 63) << 2;
    int s = src[e];
    int d = dst[e];
    float scale = rdeg_src[s];
    const float4 v = *reinterpret_cast<const float4*>(feat + (size_t)s * D_FEAT + q);
    float* ap = agg + (size_t)d * D_FEAT + q;
    atomicAdd(ap + 0, v.x * scale);
    atomicAdd(ap + 1, v.y * scale);
    atomicAdd(ap + 2, v.z * scale);
    atomicAdd(ap + 3, v.w * scale);
}

// ---------------------------------------------------------------------------
// out[m, n] = relu( (A @ W)[m, n] * rdeg[m] + bias[n] )
// A: [M, 256], W: [256, 256], M a multiple of 32.
// Block = 256 threads = 8 waves. The block's 32x256 A tile is staged in LDS
// once (shared by the 4 waves per M-tile), then each wave computes a 16x64
// strip: rows = blockIdx.x*32 + (w>>2)*16, cols = (w&3)*64.
// K-loop steps by 4 using V_WMMA_F32_16X16X4_F32; one A fragment (from LDS)
// feeds 4 accumulators for A-reuse. W streams from L2 (256KB, reused by all
// blocks).
//
// Fragment layout (ISA 7.12.2, 32-bit A 16x4): lane L holds row M=L&15,
// VGPR0/1 hold K = (L>>4)*2 and +1. B (4x16) mirrored: lane L holds col
// N=L&15, VGPR0/1 hold rows K=(L>>4)*2 and +1. C/D: VGPR i -> M = i + 8*(L>>4),
// N = L&15 (within the 16-col tile).
// ---------------------------------------------------------------------------
__global__ void __launch_bounds__(256)
gemm_norm_relu_kernel(const float* __restrict__ A, const float* __restrict__ W,
                      const float* __restrict__ bias, const float* __restrict__ rdeg,
                      float* __restrict__ out) {
    __shared__ float As[32 * LDS_STRIDE];   // 33.3 KB of the WGP's 320 KB

    const int tid  = threadIdx.x;
    const int lane = tid & 31;
    const int wave = tid >> 5;

    // --- Stage the 32x256 A tile into LDS (each thread: 8 float4 quads) ---
    {
        const float* Ablk = A + (size_t)blockIdx.x * 32 * D_FEAT;
        #pragma unroll
        for (int i = 0; i < 8; ++i) {
            int quad = tid + 256 * i;          // 0..2047
            int r    = quad >> 6;              // 0..31
            int c    = (quad & 63) << 2;       // 0..252
            float4 v = *reinterpret_cast<const float4*>(Ablk + (size_t)r * D_FEAT + c);
            *reinterpret_cast<float4*>(&As[r * LDS_STRIDE + c]) = v;  // 16B-aligned (1040B row stride)
        }
    }
    __syncthreads();

    const int row    = blockIdx.x * 32 + (wave >> 2) * 16;
    const int col    = (wave & 3) * 64;
    const int mlocal = (wave >> 2) * 16 + (lane & 15);  // LDS row this lane carries
    const int kb     = (lane >> 4) << 1;                // K sub-offset (0 or 2)
    const int ncol   = lane & 15;                       // B/D column this lane carries

    v8f acc0 = {}, acc1 = {}, acc2 = {}, acc3 = {};
    const float* Alds = &As[mlocal * LDS_STRIDE];

    #pragma unroll 4
    for (int k = 0; k < D_FEAT; k += 4) {
        v2f a;
        a.x = Alds[k + kb];          // ds_load_b64 (8B aligned: k%4==0, kb even)
        a.y = Alds[k + kb + 1];

        const float* W0p = W + (size_t)(k + kb) * D_FEAT + col + ncol;
        const float* W1p = W0p + D_FEAT;
        v2f b0; b0.x = W0p[0];  b0.y = W1p[0];
        v2f b1; b1.x = W0p[16]; b1.y = W1p[16];
        v2f b2; b2.x = W0p[32]; b2.y = W1p[32];
        v2f b3; b3.x = W0p[48]; b3.y = W1p[48];

        acc0 = __builtin_amdgcn_wmma_f32_16x16x4_f32(false, a, false, b0, (short)0, acc0, false, false);
        acc1 = __builtin_amdgcn_wmma_f32_16x16x4_f32(false, a, false, b1, (short)0, acc1, false, false);
        acc2 = __builtin_amdgcn_wmma_f32_16x16x4_f32(false, a, false, b2, (short)0, acc2, false, false);
        acc3 = __builtin_amdgcn_wmma_f32_16x16x4_f32(false, a, false, b3, (short)0, acc3, false, false);
    }

    const int half = lane >> 4;
    const int n0   = col + ncol;
    const float bb0 = bias[n0];
    const float bb1 = bias[n0 + 16];
    const float bb2 = bias[n0 + 32];
    const float bb3 = bias[n0 + 48];

    #pragma unroll
    for (int i = 0; i < 8; ++i) {
        int m = row + i + half * 8;
        float scale = rdeg[m];
        float* orow = out + (size_t)m * D_FEAT;
        orow[n0]      = fmaxf(acc0[i] * scale + bb0, 0.0f);
        orow[n0 + 16] = fmaxf(acc1[i] * scale + bb1, 0.0f);
        orow[n0 + 32] = fmaxf(acc2[i] * scale + bb2, 0.0f);
        orow[n0 + 48] = fmaxf(acc3[i] * scale + bb3, 0.0f);
    }
}

// ---------------------------------------------------------------------------
// Launch
// ---------------------------------------------------------------------------
extern "C" void kernel_launch(void* const* d_in, const int* in_sizes, int n_in,
                              void* d_out, int out_size, void* d_ws, size_t ws_size,
                              hipStream_t stream) {
    const float* x   = (const float*)d_in[0];
    const int*  src0 = (const int*)d_in[1];
    const int*  dst0 = (const int*)d_in[2];
    const int*  src1 = (const int*)d_in[3];
    const int*  dst1 = (const int*)d_in[4];
    const float* W0  = (const float*)d_in[5];
    const float* b0  = (const float*)d_in[6];
    const float* W1  = (const float*)d_in[7];
    const float* b1  = (const float*)d_in[8];

    const int N_SRC0 = in_sizes[0] / D_FEAT;   // 400000
    const int E0     = in_sizes[1];            // 400000
    const int E1     = in_sizes[3];            // 40000
    const int N_DST0 = 40000;                  // per reference setup
    const int N_DST1 = 4000;

    // Workspace layout (floats):
    //   [rdeg_out0 | rdeg_in0 | rdeg_out1 | rdeg_in1 | agg0 | h | agg1]
    float* ws        = (float*)d_ws;
    float* rdeg_out0 = ws;                                  // N_SRC0
    float* rdeg_in0  = rdeg_out0 + N_SRC0;                  // N_DST0
    float* rdeg_out1 = rdeg_in0 + N_DST0;                   // N_DST0
    float* rdeg_in1  = rdeg_out1 + N_DST0;                  // N_DST1
    float* agg0      = rdeg_in1 + N_DST1;                   // N_DST0 * 256
    float* h         = agg0 + (size_t)N_DST0 * D_FEAT;      // N_DST0 * 256
    float* agg1      = h + (size_t)N_DST0 * D_FEAT;         // N_DST1 * 256

    const size_t degN = (size_t)N_SRC0 + 2 * (size_t)N_DST0 + N_DST1;

    hipMemsetAsync(ws, 0, degN * sizeof(float), stream);
    hipMemsetAsync(agg0, 0, (size_t)N_DST0 * D_FEAT * sizeof(float), stream);
    hipMemsetAsync(agg1, 0, (size_t)N_DST1 * D_FEAT * sizeof(float), stream);

    // Degrees for both layers, then rsqrt(clip(.,1)) in one sweep.
    degree_kernel<<<(E0 + 255) / 256, 256, 0, stream>>>(src0, dst0, rdeg_out0, rdeg_in0, E0);
    degree_kernel<<<(E1 + 255) / 256, 256, 0, stream>>>(src1, dst1, rdeg_out1, rdeg_in1, E1);
    rsqrt_deg_kernel<<<((int)degN + 255) / 256, 256, 0, stream>>>(ws, (int)degN);

    // Layer 0: aggregate then GEMM (+norm, +bias, +relu)
    aggregate_kernel<<<(E0 * 64 + 255) / 256, 256, 0, stream>>>(x, src0, dst0, rdeg_out0, agg0, E0);
    gemm_norm_relu_kernel<<<N_DST0 / 32, 256, 0, stream>>>(agg0, W0, b0, rdeg_in0, h);

    // Layer 1
    aggregate_kernel<<<(E1 * 64 + 255) / 256, 256, 0, stream>>>(h, src1, dst1, rdeg_out1, agg1, E1);
    gemm_norm_relu_kernel<<<N_DST1 / 32, 256, 0, stream>>>(agg1, W1, b1, rdeg_in1, (float*)d_out);
}